// LabelSmoothing_41042707480771
// MI455X (gfx1250) — compile-verified
//
#include <hip/hip_runtime.h>
#include <hip/hip_bf16.h>
#include <math.h>
#include <stdint.h>

// Problem constants (from reference)
#define NROWS     4096
#define VDIM      50257
#define NTHREADS  256
#define TILE      7168                 // floats per LDS tile = 7 * (256 threads * float4)
#define V4PT      7                    // float4 loads per thread per full tile
#define EPT       (V4PT * 4)           // 28 elements per thread per full tile
#define LOG2E     1.4426950408889634f

typedef unsigned int u32x4 __attribute__((ext_vector_type(4)));
typedef int          i32x8 __attribute__((ext_vector_type(8)));
typedef int          i32x4 __attribute__((ext_vector_type(4)));

// Issue a 1-D TDM tensor_load: n fp32 elements from gsrc to the LDS byte address in lds_off.
// D# layout per CDNA5 ISA ch.8:
//  group0: [1:0]=count=1, [63:32]=lds_addr, [120:64]=global_addr, [127:126]=type=2
//  group1: data_size=2(4B) @ [17:16]; tensor_dim0=n; tensor_dim1=1; tile_dim0=n;
//          tile_dim1=1; stride0=n; workgroup_mask=0 (not in cluster)
__device__ __forceinline__ void tdm_load_1d(const float* gsrc, unsigned lds_off, unsigned n) {
    unsigned long long ga = (unsigned long long)(uintptr_t)gsrc;
    u32x4 g0;
    g0[0] = 1u;                                               // count=1, user mode
    g0[1] = lds_off;                                          // LDS byte address
    g0[2] = (unsigned)ga;                                     // global addr [31:0]
    g0[3] = ((unsigned)(ga >> 32) & 0x1FFFFFFu) | (2u << 30); // addr [56:32] | type=2
    i32x8 g1;
    g1[0] = 0x20000;                                  // mask=0, data_size=2 (4 bytes)
    g1[1] = (int)((n & 0xFFFFu) << 16);               // tensor_dim0[15:0] @ bits[63:48]
    g1[2] = (int)((n >> 16) | (1u << 16));            // tensor_dim0[31:16] | tensor_dim1=1
    g1[3] = (int)(n << 16);                           // tensor_dim1 hi=0 | tile_dim0=n
    g1[4] = 1;                                        // tile_dim1=1, tile_dim2=0
    g1[5] = (int)n;                                   // tensor_dim0_stride[31:0]
    g1[6] = 0;
    g1[7] = 0;
    i32x4 z4 = {0, 0, 0, 0};                          // groups 2/3 unused (<=2D)
    i32x8 z8 = {0, 0, 0, 0, 0, 0, 0, 0};              // extra group (clang-23 6-arg form)
    __builtin_amdgcn_tensor_load_to_lds(g0, g1, z4, z4, z8, 0);
}

// One block per row: streaming blockwise online softmax over TDM-fed LDS tiles.
// Writes ws[2*row]   = sum_j scores[row][j] = rowsum - V*lse
//        ws[2*row+1] = scores[row][label]   = logit_lbl - lse
__global__ __launch_bounds__(NTHREADS) void ls_row_kernel(
    const float* __restrict__ logits, const int* __restrict__ labels,
    float* __restrict__ ws) {
    __shared__ __align__(128) float buf[2][TILE];
    __shared__ float red_m[8], red_s[8], red_r[8];

    const int row = blockIdx.x;
    const int tid = threadIdx.x;
    const float* rowptr = logits + (size_t)row * VDIM;

    float lbl_logit = 0.f;
    if (tid == 0) {
        const int lbl = labels[row];
        lbl_logit = rowptr[lbl];
    }

    const int ntiles = (VDIM + TILE - 1) / TILE;   // 8 (7 full + 81-elem tail)

    // Prologue: wave0 kicks off DMA of tile 0 into buf[0]
    if (tid == 0) {
        tdm_load_1d(rowptr, (unsigned)(uintptr_t)&buf[0][0],
                    (unsigned)(VDIM < TILE ? VDIM : TILE));
    }

    float m = -INFINITY, s = 0.f, r = 0.f;
    int cur = 0;

    for (int t = 0; t < ntiles; ++t) {
        const int base = t * TILE;
        const int len  = (VDIM - base) < TILE ? (VDIM - base) : TILE;

        if (tid == 0) {
            if (t + 1 < ntiles) {
                const int nbase = base + TILE;
                const int nlen  = (VDIM - nbase) < TILE ? (VDIM - nbase) : TILE;
                // Prefetch next tile into the other buffer, then wait for
                // the CURRENT tile (TDM ops complete in order per wave).
                tdm_load_1d(rowptr + nbase, (unsigned)(uintptr_t)&buf[cur ^ 1][0],
                            (unsigned)nlen);
                __builtin_amdgcn_s_wait_tensorcnt((short)1);
            } else {
                __builtin_amdgcn_s_wait_tensorcnt((short)0);
            }
        }
        __syncthreads();   // buf[cur] is valid for all 8 waves

        float xv[EPT];
        float tm = -INFINITY;

        if (len == TILE) {
            // Fast path: ds_load_b128, 7 float4 per thread, all in VGPRs.
            const float4* p4 = (const float4*)&buf[cur][0];
#pragma unroll
            for (int k = 0; k < V4PT; ++k) {
                const float4 v = p4[tid + k * NTHREADS];
                xv[4 * k + 0] = v.x; xv[4 * k + 1] = v.y;
                xv[4 * k + 2] = v.z; xv[4 * k + 3] = v.w;
                tm = fmaxf(tm, fmaxf(fmaxf(v.x, v.y), fmaxf(v.z, v.w)));
                r += (v.x + v.y) + (v.z + v.w);
            }
        } else {
            // Tail tile: guarded scalar path.
#pragma unroll
            for (int k = 0; k < EPT; ++k) {
                const int idx = tid + k * NTHREADS;
                if (idx < len) {
                    const float v = buf[cur][idx];
                    xv[k] = v;
                    tm = fmaxf(tm, v);
                    r += v;
                } else {
                    xv[k] = -INFINITY;   // exp2(fma(-inf,...)) == 0
                }
            }
        }

        // exp pass against merged max: exp(x-nm) = exp2(fma(x, log2e, -nm*log2e))
        const float nm = fmaxf(m, tm);
        const float nmc = -nm * LOG2E;
        float ts = 0.f;
#pragma unroll
        for (int k = 0; k < EPT; ++k) ts += exp2f(fmaf(xv[k], LOG2E, nmc));
        s = s * exp2f((m - nm) * LOG2E) + ts;   // m==-inf on first tile -> 0*0 == 0
        m = nm;

        __syncthreads();   // everyone done reading buf[cur] before it is refilled
        cur ^= 1;
    }

    // Intra-wave (wave32) online-softmax tree reduction
#pragma unroll
    for (int off = 16; off > 0; off >>= 1) {
        const float om  = __shfl_down(m, off, 32);
        const float os  = __shfl_down(s, off, 32);
        const float orr = __shfl_down(r, off, 32);
        const float nm2 = fmaxf(m, om);
        s = s * exp2f((m - nm2) * LOG2E) + os * exp2f((om - nm2) * LOG2E);
        m = nm2;
        r += orr;
    }
    const int wid = tid >> 5, lane = tid & 31;
    if (lane == 0) { red_m[wid] = m; red_s[wid] = s; red_r[wid] = r; }
    __syncthreads();

    if (tid == 0) {
        float M = -INFINITY, S = 0.f, R = 0.f;
#pragma unroll
        for (int i = 0; i < 8; ++i) {
            const float nm2 = fmaxf(M, red_m[i]);
            S = S * exp2f((M - nm2) * LOG2E) + red_s[i] * exp2f((red_m[i] - nm2) * LOG2E);
            M = nm2;
            R += red_r[i];
        }
        const float lse = M + __logf(S);
        ws[2 * row]     = R - (float)VDIM * lse;   // sum_j scores[row][j]
        ws[2 * row + 1] = lbl_logit - lse;         // scores[row][label]
    }
}

// Reduce the 4096 per-row partials and apply the closed-form formula.
__global__ __launch_bounds__(256) void ls_final_kernel(
    const float* __restrict__ ws, float* __restrict__ out) {
    __shared__ double sa[256], sb[256];
    const int tid = threadIdx.x;
    double a = 0.0, b = 0.0;
    for (int i = tid; i < NROWS; i += 256) {
        a += (double)ws[2 * i];
        b += (double)ws[2 * i + 1];
    }
    sa[tid] = a; sb[tid] = b;
    __syncthreads();
    for (int st = 128; st > 0; st >>= 1) {
        if (tid < st) { sa[tid] += sa[tid + st]; sb[tid] += sb[tid + st]; }
        __syncthreads();
    }
    if (tid == 0) {
        const double V = (double)VDIM, Nr = (double)NROWS;
        const double smooth = 0.1 / (V - 1.0);
        const double conf = 1.0 - 0.1;
        const double log_smooth = log(smooth);
        const double log_conf = log(conf);
        const double S1 = sa[0];   // sum over all scores
        const double S2 = sb[0];   // sum of scores at labels
        // base + corr (see reference):
        const double res = smooth * (log_smooth * Nr * V - S1)
                         + Nr * (conf * log_conf - smooth * log_smooth)
                         - (conf - smooth) * S2;
        out[0] = (float)res;
    }
}

extern "C" void kernel_launch(void* const* d_in, const int* in_sizes, int n_in,
                              void* d_out, int out_size, void* d_ws, size_t ws_size,
                              hipStream_t stream) {
    const float* logits = (const float*)d_in[0];
    const int*   labels = (const int*)d_in[1];
    float* ws  = (float*)d_ws;    // 4096 * 2 floats = 32 KB of scratch
    float* out = (float*)d_out;

    ls_row_kernel<<<NROWS, NTHREADS, 0, stream>>>(logits, labels, ws);
    ls_final_kernel<<<1, 256, 0, stream>>>(ws, out);
}